// ST_OHKW_MSELoss_89249420411523
// MI455X (gfx1250) — compile-verified
//
#include <hip/hip_runtime.h>

// Problem constants (match reference)
#define NB   64
#define NJ   17
#define NHW  9216      // 96*96
#define NTOP 8
#define NTHREADS 256
#define NITER (NHW / 4 / NTHREADS)   // 2304 float4 / 256 threads = 9 exactly

typedef __attribute__((ext_vector_type(2))) float v2f;
typedef __attribute__((ext_vector_type(4))) float vf4;
typedef __attribute__((ext_vector_type(8))) float v8f;

// ---------------------------------------------------------------------------
// Pass 1: one block per (b,j). Streams s,t,g with NT b128 loads (fully
// unrolled, 27 independent loads -> s_clause batching / deep LOADcnt
// pipelining), accumulates
//   psg = sum (s-g)^2, pst = sum (s-t)^2, pmax = max g   over HW pixels,
// then scales by w^2 and reduces across the block. Final 8 wave-partials are
// collapsed with a single V_WMMA_F32_16X16X4_F32 (ones/selector B matrix).
// ---------------------------------------------------------------------------
__global__ __launch_bounds__(NTHREADS) void st_ohkm_pass1(
    const float* __restrict__ s, const float* __restrict__ t,
    const float* __restrict__ g, const float* __restrict__ w,
    float* __restrict__ sum_sg, float* __restrict__ sum_st,
    float* __restrict__ gmax)
{
    const int bj = blockIdx.x;                 // flattened (b*NJ + j)
    const int tid = threadIdx.x;
    const size_t base = (size_t)bj * NHW;

    const vf4* s4 = (const vf4*)(s + base);
    const vf4* t4 = (const vf4*)(t + base);
    const vf4* g4 = (const vf4*)(g + base);

    float psg = 0.0f, pst = 0.0f, pmax = -__builtin_inff();
#pragma unroll
    for (int k = 0; k < NITER; ++k) {
        const int i = tid + k * NTHREADS;
        vf4 sv = __builtin_nontemporal_load(&s4[i]);   // th:NT, read-once stream
        vf4 tv = __builtin_nontemporal_load(&t4[i]);
        vf4 gv = __builtin_nontemporal_load(&g4[i]);
        vf4 dsg = sv - gv; dsg *= dsg;
        vf4 dst = sv - tv; dst *= dst;
        psg += (dsg.x + dsg.y) + (dsg.z + dsg.w);
        pst += (dst.x + dst.y) + (dst.z + dst.w);
        pmax = fmaxf(pmax, fmaxf(fmaxf(gv.x, gv.y), fmaxf(gv.z, gv.w)));
    }
    const float wj = w[bj];
    const float w2 = wj * wj;                  // (s*w - g*w)^2 == w^2 (s-g)^2
    psg *= w2;
    pst *= w2;

    // wave32 tree reduction
    for (int off = 16; off >= 1; off >>= 1) {
        psg += __shfl_xor(psg, off, 32);
        pst += __shfl_xor(pst, off, 32);
        pmax = fmaxf(pmax, __shfl_xor(pmax, off, 32));
    }

    __shared__ float lsg[8], lst[8], lmx[8];
    if ((tid & 31) == 0) {
        lsg[tid >> 5] = psg;
        lst[tid >> 5] = pst;
        lmx[tid >> 5] = pmax;
    }
    __syncthreads();

    if (tid < 32) {                            // wave 0, EXEC all ones
        const int lane = tid;
        // A (16x4 f32): lane m<8 holds A[m,0]=wave_sg[m], A[m,1]=wave_st[m]
        v2f a; a.x = 0.0f; a.y = 0.0f;
        if (lane < 8) { a.x = lsg[lane]; a.y = lst[lane]; }
        // B (4x16) selector: cols 0-7 pick even K (sg), cols 8-15 odd K (st)
        const int n = lane & 15;
        v2f b; b.x = (n < 8) ? 1.0f : 0.0f;
        b.y = (n < 8) ? 0.0f : 1.0f;
        v8f c = {};
        v8f d = __builtin_amdgcn_wmma_f32_16x16x4_f32(
            /*neg_a=*/false, a, /*neg_b=*/false, b,
            /*c_mod=*/(short)0, c, /*reuse_a=*/false, /*reuse_b=*/false);
        // lane 0 holds D[0..7,0] = sg row-sums; lane 8 holds D[0..7,8] = st
        float local = ((d[0] + d[1]) + (d[2] + d[3])) +
                      ((d[4] + d[5]) + (d[6] + d[7]));
        const float tot_sg = __shfl(local, 0, 32);
        const float tot_st = __shfl(local, 8, 32);

        float m = (lane < 8) ? lmx[lane] : -__builtin_inff();
        for (int off = 4; off >= 1; off >>= 1)
            m = fmaxf(m, __shfl_xor(m, off, 32));

        if (lane == 0) {
            sum_sg[bj] = tot_sg;
            sum_st[bj] = tot_st;
            gmax[bj]   = m;
        }
    }
}

// ---------------------------------------------------------------------------
// Pass 2: single tiny block. cond[j], loss_mat, per-sample top-8/17 (OHKM),
// joint-summed MSE, final three scalars.
// ---------------------------------------------------------------------------
__global__ __launch_bounds__(128) void st_ohkm_pass2(
    const float* __restrict__ sum_sg, const float* __restrict__ sum_st,
    const float* __restrict__ gmax, const int* __restrict__ enj,
    float* __restrict__ out)
{
    __shared__ float cond[NJ];
    __shared__ float mse_j[NJ];
    __shared__ float scratch[NB * NJ];
    __shared__ float ohkm_b[NB];
    const int tid = threadIdx.x;

    if (tid < NJ) {
        float mx = -__builtin_inff();
        float acc_sg = 0.0f, acc_st = 0.0f;
        for (int b = 0; b < NB; ++b) {
            const int i = b * NJ + tid;
            mx = fmaxf(mx, gmax[i]);
            acc_sg += sum_sg[i];
            acc_st += sum_st[i];
        }
        const float cj = (mx == 1.0f) ? 1.0f : 0.0f;
        cond[tid] = cj;
        const float inv = 1.0f / (float)(NB * NHW);
        const float msg = acc_sg * inv;
        const float mst = acc_st * inv;
        mse_j[tid] = (cj != 0.0f) ? msg : (msg + mst);
    }
    __syncthreads();

    for (int i = tid; i < NB * NJ; i += 128) {
        const int j = i % NJ;
        const float ssum = (cond[j] != 0.0f) ? sum_sg[i] : (sum_sg[i] + sum_st[i]);
        scratch[i] = 0.5f * ssum * (1.0f / (float)NHW);   // loss_mat[b,j]
    }
    __syncthreads();

    if (tid < NB) {
        float acc = 0.0f;
        for (int k = 0; k < NTOP; ++k) {                  // top-8 of 17
            float best = -__builtin_inff();
            int bi = 0;
            for (int j = 0; j < NJ; ++j) {
                const float v = scratch[tid * NJ + j];
                if (v > best) { best = v; bi = j; }
            }
            acc += best;
            scratch[tid * NJ + bi] = -__builtin_inff();
        }
        ohkm_b[tid] = acc * (1.0f / (float)NTOP);
    }
    __syncthreads();

    if (tid == 0) {
        float oh = 0.0f;
        for (int b = 0; b < NB; ++b) oh += ohkm_b[b];
        oh *= (1.0f / (float)NB);
        float mse = 0.0f;
        for (int j = 0; j < NJ; ++j) mse += mse_j[j];
        out[0] = oh;
        out[1] = mse / (float)enj[0];
        out[2] = oh + mse;
    }
}

extern "C" void kernel_launch(void* const* d_in, const int* in_sizes, int n_in,
                              void* d_out, int out_size, void* d_ws, size_t ws_size,
                              hipStream_t stream) {
    const float* s = (const float*)d_in[0];       // output_s [B,J,H,W]
    const float* t = (const float*)d_in[1];       // output_t
    const float* g = (const float*)d_in[2];       // target
    const float* w = (const float*)d_in[3];       // target_weight [B,J,1]
    const int* enj = (const int*)d_in[4];         // effective_num_joints

    float* ws = (float*)d_ws;
    float* sum_sg = ws;                           // [B*J]
    float* sum_st = ws + NB * NJ;                 // [B*J]
    float* gmx    = ws + 2 * NB * NJ;             // [B*J]
    float* out = (float*)d_out;                   // 3 scalars

    st_ohkm_pass1<<<NB * NJ, NTHREADS, 0, stream>>>(s, t, g, w, sum_sg, sum_st, gmx);
    st_ohkm_pass2<<<1, 128, 0, stream>>>(sum_sg, sum_st, gmx, enj, out);
}